// Undefine_loss_33818572488961
// MI455X (gfx1250) — compile-verified
//
#include <hip/hip_runtime.h>
#include <math.h>

typedef __attribute__((ext_vector_type(2))) float v2f;
typedef __attribute__((ext_vector_type(4))) float v4f;
typedef __attribute__((ext_vector_type(8))) float v8f;

// Exact wave32 sum using V_WMMA_F32_16X16X4_F32 with A == all-ones.
// B vgpr0 carries the 32 per-lane partials (vgpr1 = 0). With A all-ones every
// row of D equals the column-sums of B; D's first VGPR holds rows {0,8}, i.e.
// each column value appears exactly twice across the 32 lanes. Summing d[0]
// over all lanes therefore gives 2 * (sum of all B elements) = 2 * wave_sum.
__device__ __forceinline__ float wave_sum32(float x) {
    v2f a; a.x = 1.0f; a.y = 1.0f;   // A matrix: all ones (16x4)
    v2f b; b.x = x;    b.y = 0.0f;   // B matrix: partials in vgpr0, rest zero
    v8f c = {};
    c = __builtin_amdgcn_wmma_f32_16x16x4_f32(
        /*neg_a=*/false, a, /*neg_b=*/false, b,
        /*c_mod=*/(short)0, c, /*reuse_a=*/false, /*reuse_b=*/false);
    float s = c[0];
#pragma unroll
    for (int off = 16; off > 0; off >>= 1)
        s += __shfl_xor(s, off, 32);
    return 0.5f * s;
}

__global__ __launch_bounds__(256)
void proj_loss_kernel(const float* __restrict__ Vt,
                      const float* __restrict__ Rt,   // [B,12,3,3] f32
                      float* __restrict__ blockSums,
                      int n,
                      float c0x, float c0y, float c1x, float c1y,
                      float c2x, float c2y, float c3x, float c3y)
{
    const float cax[4] = {c0x, c1x, c2x, c3x};
    const float cay[4] = {c0y, c1y, c2y, c3y};
    const float invW = 1.0f / 1920.0f;
    const float invH = 1.0f / 1080.0f;

    float partial = 0.0f;
    const int stride = gridDim.x * blockDim.x;
    for (int i = blockIdx.x * blockDim.x + threadIdx.x; i < n; i += stride) {
        // ---- quaternion -> rotation (no sqrt: all terms quadratic) ----
        // Single-use 450+MB stream >> 192MB L2: use non-temporal loads.
        v4f q = __builtin_nontemporal_load((const v4f*)Vt + i);
        float x = q.x, y = q.y, z = q.z, w = q.w;
        float s2 = 2.0f / (x*x + y*y + z*z + w*w);
        float xx = s2*x*x, yy = s2*y*y, zz = s2*z*z;
        float xy = s2*x*y, xz = s2*x*z, yz = s2*y*z;
        float xw = s2*x*w, yw = s2*y*w, zw = s2*z*w;
        float r00 = 1.0f - (yy + zz), r01 = xy - zw,          r02 = xz + yw;
        float r10 = xy + zw,          r11 = 1.0f - (xx + zz), r12 = yz - xw;
        float r20 = xz - yw,          r21 = yz + xw,          r22 = 1.0f - (xx + yy);

        // ---- Pr = real_projection_t[i, 6] : 9 floats at i*108 + 54 ----
        // Base offset 216B + i*432B is always 8B-aligned: 4x b64 + 1x b32, NT.
        const float* Pbase = Rt + (size_t)i * 108 + 54;
        const v2f* P2 = (const v2f*)Pbase;
        v2f ab = __builtin_nontemporal_load(P2 + 0);
        v2f cd = __builtin_nontemporal_load(P2 + 1);
        v2f ef = __builtin_nontemporal_load(P2 + 2);
        v2f gh = __builtin_nontemporal_load(P2 + 3);
        float p22 = __builtin_nontemporal_load(Pbase + 8);
        float p00 = ab.x, p01 = ab.y, p02 = cd.x;
        float p10 = cd.y, p11 = ef.x, p12 = ef.y;
        float p20 = gh.x, p21 = gh.y;

        // ---- per corner: p = Pr * (R^T * (K^-1 corner)), hinge violation ----
        float best = 0.0f;
#pragma unroll
        for (int cI = 0; cI < 4; ++cI) {
            float a  = cax[cI];
            float bb = cay[cI];
            // u = R^T * (a, bb, 1)
            float u0 = r00*a + r10*bb + r20;
            float u1 = r01*a + r11*bb + r21;
            float u2 = r02*a + r12*bb + r22;
            // p = Pr * u
            float px = p00*u0 + p01*u1 + p02*u2;
            float py = p10*u0 + p11*u1 + p12*u2;
            float pz = p20*u0 + p21*u1 + p22*u2;
            float invz = 1.0f / pz;
            float xn = px * invz * invW;
            float yn = py * invz * invH;
            float v = fmaxf(-xn, 0.0f) + fmaxf(xn - 1.0f, 0.0f)
                    + fmaxf(-yn, 0.0f) + fmaxf(yn - 1.0f, 0.0f);
            best = fmaxf(best, v);
        }
        partial += best;
    }

    // ---- deterministic block reduction: WMMA wave sum -> LDS -> one value ----
    float wsum = wave_sum32(partial);
    __shared__ float sm[8];
    int lane = threadIdx.x & 31;
    int wv   = threadIdx.x >> 5;
    if (lane == 0) sm[wv] = wsum;
    __syncthreads();
    if (threadIdx.x == 0) {
        float t = 0.0f;
#pragma unroll
        for (int k = 0; k < 8; ++k) t += sm[k];
        blockSums[blockIdx.x] = t;
    }
}

__global__ __launch_bounds__(256)
void final_reduce_kernel(const float* __restrict__ blockSums, int nblocks,
                         float* __restrict__ out)
{
    float p = 0.0f;
    for (int i = threadIdx.x; i < nblocks; i += 256) p += blockSums[i];
    float wsum = wave_sum32(p);
    __shared__ float sm[8];
    int lane = threadIdx.x & 31;
    int wv   = threadIdx.x >> 5;
    if (lane == 0) sm[wv] = wsum;
    __syncthreads();
    if (threadIdx.x == 0) {
        float t = 0.0f;
#pragma unroll
        for (int k = 0; k < 8; ++k) t += sm[k];
        out[0] = t;
    }
}

extern "C" void kernel_launch(void* const* d_in, const int* in_sizes, int n_in,
                              void* d_out, int out_size, void* d_ws, size_t ws_size,
                              hipStream_t stream) {
    const float* Vt = (const float*)d_in[0];          // [B,4]
    const float* Rt = (const float*)d_in[1];          // [B,12,3,3]
    float* out      = (float*)d_out;                  // [1]
    float* bsum     = (float*)d_ws;                   // per-block partials

    int n = in_sizes[0] / 4;                          // B
    const int threads = 256;
    int blocks = (n + threads * 4 - 1) / (threads * 4);
    if (blocks > 1024) blocks = 1024;
    if (blocks < 1)    blocks = 1;

    // Host-side constants (double precision): f and K^{-1}-transformed corners.
    double f  = 0.5 * 1080.0 / tan(0.5 * 0.8);
    double cx = 0.5 * 1920.0, cy = 0.5 * 1080.0;
    double x0 = (double)(int)(1920.0 * 0.1), x1 = (double)(int)(1920.0 * 0.9);
    double y0 = (double)(int)(1080.0 * 0.1), y1 = (double)(int)(1080.0 * 0.9);
    float c0x = (float)((x0 - cx) / f), c0y = (float)((y0 - cy) / f);
    float c1x = (float)((x0 - cx) / f), c1y = (float)((y1 - cy) / f);
    float c2x = (float)((x1 - cx) / f), c2y = (float)((y0 - cy) / f);
    float c3x = (float)((x1 - cx) / f), c3y = (float)((y1 - cy) / f);

    proj_loss_kernel<<<blocks, threads, 0, stream>>>(
        Vt, Rt, bsum, n, c0x, c0y, c1x, c1y, c2x, c2y, c3x, c3y);
    final_reduce_kernel<<<1, threads, 0, stream>>>(bsum, blocks, out);
}